// Sgc_83330955477196
// MI455X (gfx1250) — compile-verified
//
#include <hip/hip_runtime.h>
#include <math.h>

typedef __attribute__((ext_vector_type(2))) float v2f;
typedef __attribute__((ext_vector_type(8))) float v8f;

#define DFEAT 128

// Hardware fp32 atomic add: unsafeAtomicAdd maps to global_atomic_add_f32 on
// AMD targets (no CAS expansion). The scatter phase issues ~205M of these,
// so the hardware fadd path is mandatory for the L2-atomic roofline.
__device__ __forceinline__ void atomic_add_f32_hw(float* addr, float v) {
    unsafeAtomicAdd(addr, v);
}

// ---------------------------------------------------------------------------
// Phase 1: degree (with self-loop) -> dinv = deg^-0.5 (in place)
// ---------------------------------------------------------------------------
__global__ void sgc_init_deg(float* __restrict__ deg, int n) {
    int i = blockIdx.x * blockDim.x + threadIdx.x;
    if (i < n) deg[i] = 1.0f;  // self-loop contributes 1 to every node's in-degree
}

__global__ void sgc_count_deg(const int* __restrict__ col, float* __restrict__ deg, int e) {
    int i = blockIdx.x * blockDim.x + threadIdx.x;
    if (i < e) {
        atomic_add_f32_hw(&deg[col[i]], 1.0f);
    }
}

__global__ void sgc_dinv(float* __restrict__ deg, int n) {
    int i = blockIdx.x * blockDim.x + threadIdx.x;
    if (i < n) deg[i] = rsqrtf(deg[i]);  // deg >= 1 always (self-loop)
}

// ---------------------------------------------------------------------------
// Phase 2: agg init with self-loop term: agg[i][:] = dinv[i]^2 * x[i][:]
// ---------------------------------------------------------------------------
__global__ void sgc_agg_init(const float* __restrict__ x, const float* __restrict__ dinv,
                             float* __restrict__ agg, int nd) {
    int idx = blockIdx.x * blockDim.x + threadIdx.x;
    if (idx < nd) {
        int node = idx >> 7;  // DFEAT == 128
        float di = dinv[node];
        agg[idx] = di * di * x[idx];
    }
}

// ---------------------------------------------------------------------------
// Phase 3: edge scatter  agg[col] += dinv[row]*dinv[col] * x[row]
// 32 threads per edge, 4 features per thread (float4 gather + 4 fp32 atomics).
// x/agg are L2-resident (51 MB each vs 192 MB L2), so this is L2-bandwidth/
// atomic-throughput bound, not HBM bound.
// ---------------------------------------------------------------------------
__global__ void sgc_scatter(const float* __restrict__ x,
                            const int* __restrict__ row, const int* __restrict__ col,
                            const float* __restrict__ dinv, float* __restrict__ agg,
                            int e) {
    int idx = blockIdx.x * blockDim.x + threadIdx.x;
    int total = e * 32;
    if (idx >= total) return;
    int eidx = idx >> 5;
    int q    = idx & 31;
    int r = row[eidx];
    int c = col[eidx];
    float norm = dinv[r] * dinv[c];
    const float4 xv = *(const float4*)(x + (size_t)r * DFEAT + q * 4);
    float* dst = agg + (size_t)c * DFEAT + q * 4;
    atomic_add_f32_hw(dst + 0, norm * xv.x);
    atomic_add_f32_hw(dst + 1, norm * xv.y);
    atomic_add_f32_hw(dst + 2, norm * xv.z);
    atomic_add_f32_hw(dst + 3, norm * xv.w);
}

// ---------------------------------------------------------------------------
// Phase 4: out = x + relu(agg @ W^T + b) using fp32 WMMA (V_WMMA_F32_16X16X4_F32).
// One wave per 16x16 output tile; block = 8 waves = one 16-node row block
// covering all 8 feature tiles (D=128). W is staged into LDS pair-interleaved:
//   Wp[k/2][n][2]  so the B operand (B[k][n] = W[n][k]) is a ds_load_b64 with
// lane-consecutive addresses (conflict friendly).
//
// A 16x4 f32 layout (per ISA): lane = half*16 + m;  a = {A[m][2*half], A[m][2*half+1]}
// B 4x16 f32 layout:           lane = half*16 + nc; b = {B[2*half][nc], B[2*half+1][nc]}
// C/D 16x16 f32 layout:        vgpr r, lane -> M = r + 8*half, N = lane&15
// ---------------------------------------------------------------------------
__global__ void __launch_bounds__(256)
sgc_wmma_gemm(const float* __restrict__ agg, const float* __restrict__ W,
              const float* __restrict__ bias, const float* __restrict__ x,
              float* __restrict__ out, int n) {
    __shared__ float Wp[DFEAT * DFEAT];  // 64 KB, pair-interleaved

    const int tid = threadIdx.y * 32 + threadIdx.x;
    // Stage W: Wp[kp*256 + nc*2 + (k&1)] = W[nc][2*kp + (k&1)]
    for (int p = tid; p < DFEAT * (DFEAT / 2); p += 256) {
        int nc = p >> 6;   // output feature 0..127
        int kp = p & 63;   // k pair 0..63
        v2f w2 = *(const v2f*)(W + (size_t)nc * DFEAT + kp * 2);
        Wp[kp * 256 + nc * 2 + 0] = w2.x;
        Wp[kp * 256 + nc * 2 + 1] = w2.y;
    }
    __syncthreads();

    const int lane  = threadIdx.x;
    const int wv    = threadIdx.y;     // 0..7 -> N tile
    const int half  = lane >> 4;       // 0 or 1
    const int l16   = lane & 15;
    const int mBase = blockIdx.x * 16;

    int m = mBase + l16;
    if (m >= n) m = n - 1;             // tail clamp (loads only; stores guarded)
    const int ncol = wv * 16 + l16;

    v8f acc = {};
    const float* arow = agg + (size_t)m * DFEAT + 2 * half;
    for (int k0 = 0; k0 < DFEAT; k0 += 4) {
        v2f a = *(const v2f*)(arow + k0);
        v2f b = *(const v2f*)(&Wp[(k0 / 2 + half) * 256 + ncol * 2]);
        acc = __builtin_amdgcn_wmma_f32_16x16x4_f32(
            /*neg_a=*/false, a, /*neg_b=*/false, b,
            /*c_mod=*/(short)0, acc, /*reuse_a=*/false, /*reuse_b=*/false);
    }

    const float bv = bias[ncol];
    for (int r = 0; r < 8; ++r) {
        int node = mBase + r + 8 * half;
        if (node < n) {
            float h = acc[r] + bv;
            h = h > 0.0f ? h : 0.0f;
            size_t o = (size_t)node * DFEAT + ncol;
            out[o] = x[o] + h;
        }
    }
}

// ---------------------------------------------------------------------------
extern "C" void kernel_launch(void* const* d_in, const int* in_sizes, int n_in,
                              void* d_out, int out_size, void* d_ws, size_t ws_size,
                              hipStream_t stream) {
    const float* x  = (const float*)d_in[0];
    const int*   ei = (const int*)d_in[1];
    const float* W  = (const float*)d_in[2];
    const float* b  = (const float*)d_in[3];
    float* out = (float*)d_out;

    const int n = in_sizes[0] / DFEAT;
    const int e = in_sizes[1] / 2;
    const int* row = ei;        // edge_index[0] = source
    const int* col = ei + e;    // edge_index[1] = target

    // Workspace: dinv[n] (deg in place) then agg[n*128]; ~51.6 MB total.
    float* deg = (float*)d_ws;
    float* agg = deg + (((size_t)n + 255) & ~(size_t)255);

    sgc_init_deg<<<(n + 255) / 256, 256, 0, stream>>>(deg, n);
    sgc_count_deg<<<(e + 255) / 256, 256, 0, stream>>>(col, deg, e);
    sgc_dinv<<<(n + 255) / 256, 256, 0, stream>>>(deg, n);

    const int nd = n * DFEAT;
    sgc_agg_init<<<(nd + 255) / 256, 256, 0, stream>>>(x, deg, agg, nd);

    const int tot = e * 32;
    sgc_scatter<<<(tot + 255) / 256, 256, 0, stream>>>(x, row, col, deg, agg, e);

    const int mTiles = (n + 15) / 16;
    sgc_wmma_gemm<<<mTiles, dim3(32, 8), 0, stream>>>(agg, W, b, x, out, n);
}